// cheby_conv_ds_42657615184278
// MI455X (gfx1250) — compile-verified
//
#include <hip/hip_runtime.h>

typedef float v2f __attribute__((ext_vector_type(2)));
typedef float v8f __attribute__((ext_vector_type(8)));

// D = A(16x16x4) x B + C in fp32 via CDNA5 WMMA
// args: (neg_a, A, neg_b, B, c_mod, C, reuse_a, reuse_b)
#define WMMA_F32(a, b, c) \
  __builtin_amdgcn_wmma_f32_16x16x4_f32(false, (a), false, (b), (short)0, (c), false, false)

// ---- CDNA5 async global->LDS path (ASYNCcnt), guarded so we never regress ----
#if defined(__gfx1250__) && __has_builtin(__builtin_amdgcn_global_load_async_to_lds_b128)
#define USE_ASYNC_LDS 1
#else
#define USE_ASYNC_LDS 0
#endif

#if USE_ASYNC_LDS
typedef int v4i __attribute__((vector_size(16)));
typedef __attribute__((address_space(1))) v4i* as1_v4ip;
typedef __attribute__((address_space(3))) v4i* as3_v4ip;

__device__ __forceinline__ void async_cp16(const float* g, float* l) {
  // generic LDS pointer: low 32 bits are the LDS offset (aperture in high bits)
  __builtin_amdgcn_global_load_async_to_lds_b128(
      (as1_v4ip)(uintptr_t)(g), (as3_v4ip)(uint32_t)(uintptr_t)(l), 0, 0);
}
__device__ __forceinline__ void wait_async_le4() {
#if __has_builtin(__builtin_amdgcn_s_wait_asynccnt)
  __builtin_amdgcn_s_wait_asynccnt(4);
#else
  asm volatile("s_wait_asynccnt 0x4" ::: "memory");
#endif
}
__device__ __forceinline__ void wait_async_le0() {
#if __has_builtin(__builtin_amdgcn_s_wait_asynccnt)
  __builtin_amdgcn_s_wait_asynccnt(0);
#else
  asm volatile("s_wait_asynccnt 0x0" ::: "memory");
#endif
}
#endif

// ---------------------------------------------------------------------------
// Y = adj @ B  (mode 0)   or   Y = 2*(adj @ B) - X  (mode 1)
// adj: [16,512,512], B/X/Y: [16,512,2048]
// Block tile 128x128, BK=16, 256 threads (8 waves as 2x4), wave tile 64x32.
// Double-buffered LDS fed by GLOBAL_LOAD_ASYNC_TO_LDS_B128.
// ---------------------------------------------------------------------------
__global__ __launch_bounds__(256) void cheby_gemm512(
    const float* __restrict__ A,   // adjacency [16,512,512]
    const float* __restrict__ Bm,  // rhs       [16,512,2048]
    const float* __restrict__ Xs,  // subtract source (mode 1), else unused
    float* __restrict__ D,         // result    [16,512,2048]
    int mode)
{
  constexpr int K = 512, N = 2048;
  constexpr int AS = 20;   // A tile (128x16) LDS row stride: 16B-aligned rows, bank-rotating
  constexpr int BS = 132;  // B tile (16x128) LDS row stride: 16B-aligned rows, bank-rotating
  __shared__ alignas(16) float Asb[2][128 * AS];
  __shared__ alignas(16) float Bsb[2][16 * BS];

  const int tid  = threadIdx.x;
  const int lane = tid & 31;
  const int wid  = tid >> 5;
  const int h    = lane >> 4;   // lane-half: 0/1
  const int l15  = lane & 15;

  const int bn = blockIdx.x * 128;
  const int bm = blockIdx.y * 128;
  const int bb = blockIdx.z;

  const float* Ab = A  + (size_t)bb * 512 * K;
  const float* Bb = Bm + (size_t)bb * K * N;

  const int wm = (wid >> 2) * 64;  // 0 or 64
  const int wn = (wid & 3) * 32;   // 0,32,64,96

  // per-thread staging coordinates (4 x 16B per thread per stage)
  const int ar = tid >> 2;         // A rows: ar, ar+64
  const int ac = (tid & 3) * 4;
  const int br = tid >> 5;         // B rows: br, br+8
  const int bc = (tid & 31) * 4;

  v8f acc[4][2];
#pragma unroll
  for (int m = 0; m < 4; ++m)
#pragma unroll
    for (int n = 0; n < 2; ++n)
      acc[m][n] = (v8f){0.f,0.f,0.f,0.f,0.f,0.f,0.f,0.f};

#if USE_ASYNC_LDS
  auto issue_stage = [&](int kt, int buf) {
    async_cp16(Ab + (size_t)(bm + ar)      * K + kt + ac, &Asb[buf][ar * AS + ac]);
    async_cp16(Ab + (size_t)(bm + ar + 64) * K + kt + ac, &Asb[buf][(ar + 64) * AS + ac]);
    async_cp16(Bb + (size_t)(kt + br)     * N + bn + bc,  &Bsb[buf][br * BS + bc]);
    async_cp16(Bb + (size_t)(kt + br + 8) * N + bn + bc,  &Bsb[buf][(br + 8) * BS + bc]);
  };
  issue_stage(0, 0);
#endif

  for (int kt = 0; kt < K; kt += 16) {
    const int cur = (kt >> 4) & 1;
#if USE_ASYNC_LDS
    if (kt + 16 < K) {
      issue_stage(kt + 16, cur ^ 1);  // prefetch next stage into other buffer
      wait_async_le4();               // in-order completion => current stage done
    } else {
      wait_async_le0();
    }
#else
    // synchronous fallback staging
    {
      const float4 va0 = *(const float4*)(Ab + (size_t)(bm + ar) * K + kt + ac);
      const float4 va1 = *(const float4*)(Ab + (size_t)(bm + ar + 64) * K + kt + ac);
      const float4 vb0 = *(const float4*)(Bb + (size_t)(kt + br) * N + bn + bc);
      const float4 vb1 = *(const float4*)(Bb + (size_t)(kt + br + 8) * N + bn + bc);
      float* a0 = &Asb[cur][ar * AS + ac];
      a0[0] = va0.x; a0[1] = va0.y; a0[2] = va0.z; a0[3] = va0.w;
      float* a1 = &Asb[cur][(ar + 64) * AS + ac];
      a1[0] = va1.x; a1[1] = va1.y; a1[2] = va1.z; a1[3] = va1.w;
      *(float4*)(&Bsb[cur][br * BS + bc]) = vb0;
      *(float4*)(&Bsb[cur][(br + 8) * BS + bc]) = vb1;
    }
#endif
    __syncthreads();

    const float* Acur = Asb[cur];
    const float* Bcur = Bsb[cur];
#pragma unroll
    for (int s = 0; s < 4; ++s) {
      const int kb = 4 * s + 2 * h;  // this lane's K pair: kb, kb+1
      v2f af[4], bf[2];
#pragma unroll
      for (int m = 0; m < 4; ++m) {
        const float* p = &Acur[(wm + m * 16 + l15) * AS + kb];
        af[m].x = p[0]; af[m].y = p[1];
      }
#pragma unroll
      for (int n = 0; n < 2; ++n) {
        const float* p = &Bcur[kb * BS + wn + n * 16 + l15];
        bf[n].x = p[0]; bf[n].y = p[BS];
      }
#pragma unroll
      for (int m = 0; m < 4; ++m)
#pragma unroll
        for (int n = 0; n < 2; ++n)
          acc[m][n] = WMMA_F32(af[m], bf[n], acc[m][n]);
    }
    __syncthreads();
  }

  // ---- epilogue: element (vgpr v, lane) = C[v + 8*h][l15] within 16x16 tile
  const size_t obase = (size_t)bb * 512 * N;
  if (mode == 0) {
#pragma unroll
    for (int m = 0; m < 4; ++m)
#pragma unroll
      for (int n = 0; n < 2; ++n) {
        const int col = bn + wn + n * 16 + l15;
#pragma unroll
        for (int v = 0; v < 8; ++v) {
          const int row = bm + wm + m * 16 + 8 * h + v;
          D[obase + (size_t)row * N + col] = acc[m][n][v];
        }
      }
  } else {
#pragma unroll
    for (int m = 0; m < 4; ++m)
#pragma unroll
      for (int n = 0; n < 2; ++n) {
        const int col = bn + wn + n * 16 + l15;
#pragma unroll
        for (int v = 0; v < 8; ++v) {
          const int row = bm + wm + m * 16 + 8 * h + v;
          const size_t idx = obase + (size_t)row * N + col;
          D[idx] = 2.0f * acc[m][n][v] - Xs[idx];
        }
      }
  }
}

// ---------------------------------------------------------------------------
// out[b,o,q,l] = bias[o] + sum_{r=0..95} Wr[o,r] * Ysrc(r)[b,q, c*64+l]
// with r = k*32 + c  (k = r>>5 selects X / Y1 / Y2; uniform per 4-row k-step)
// Wr[o,r] = W[o, c*3+k].  Block: 64 x 128 output tile, 8 waves, wave = 64x16.
// ---------------------------------------------------------------------------
__global__ __launch_bounds__(256) void cheby_mix(
    const float* __restrict__ X,    // [16,512,2048]
    const float* __restrict__ Y1,   // [16,512,2048]
    const float* __restrict__ Y2,   // [16,512,2048]
    const float* __restrict__ W,    // [64,96]
    const float* __restrict__ bias, // [64]
    float* __restrict__ out)        // [16,64,512,64]
{
  constexpr int WS = 98;  // padded LDS row stride for rearranged W (64 x 96)
  __shared__ float Wl[64 * WS];
  __shared__ float bl[64];

  const int tid  = threadIdx.x;
  const int lane = tid & 31;
  const int wid  = tid >> 5;
  const int h    = lane >> 4;
  const int l15  = lane & 15;

  const int bn = blockIdx.x * 128;  // column block within [0, 32768)
  const int bb = blockIdx.y;        // batch

  // stage rearranged W: Wl[o][r] = W[o][(r&31)*3 + (r>>5)]
  for (int i = tid; i < 64 * 96; i += 256) {
    const int o = i / 96, r = i - o * 96;
    Wl[o * WS + r] = W[o * 96 + (r & 31) * 3 + (r >> 5)];
  }
  if (tid < 64) bl[tid] = bias[tid];
  __syncthreads();

  // this wave's 16-column strip; col = q*64 + l
  const int col = bn + wid * 16 + l15;
  const int q   = col >> 6;
  const int li  = col & 63;

  const float* bases[3] = {X, Y1, Y2};

  v8f acc[4];
#pragma unroll
  for (int m = 0; m < 4; ++m)
    acc[m] = (v8f){0.f,0.f,0.f,0.f,0.f,0.f,0.f,0.f};

#pragma unroll
  for (int s = 0; s < 24; ++s) {
    const int kb   = 4 * s;
    const int kidx = kb >> 5;             // 0,1,2 : uniform per step
    const int c0   = (kb & 31) + 2 * h;   // this lane's c pair: c0, c0+1
    const float* bp = bases[kidx] + (size_t)(bb * 512 + q) * 2048 + li;
    v2f bf;
    bf.x = bp[(size_t)c0 * 64];
    bf.y = bp[(size_t)(c0 + 1) * 64];
#pragma unroll
    for (int m = 0; m < 4; ++m) {
      const float* p = &Wl[(m * 16 + l15) * WS + kb + 2 * h];
      v2f af; af.x = p[0]; af.y = p[1];
      acc[m] = WMMA_F32(af, bf, acc[m]);
    }
  }

#pragma unroll
  for (int m = 0; m < 4; ++m)
#pragma unroll
    for (int v = 0; v < 8; ++v) {
      const int o = m * 16 + 8 * h + v;
      out[(((size_t)bb * 64 + o) * 512 + q) * 64 + li] = acc[m][v] + bl[o];
    }
}

// ---------------------------------------------------------------------------
extern "C" void kernel_launch(void* const* d_in, const int* in_sizes, int n_in,
                              void* d_out, int out_size, void* d_ws, size_t ws_size,
                              hipStream_t stream) {
  const float* x    = (const float*)d_in[0];  // [16,512,32,64] == [16,512,2048]
  const float* adj  = (const float*)d_in[1];  // [16,512,512]
  const float* W    = (const float*)d_in[2];  // [64,96]
  const float* bias = (const float*)d_in[3];  // [64]
  float* out = (float*)d_out;                 // [16,64,512,64]

  const size_t ysz = (size_t)16 * 512 * 2048; // 16.7M floats = 64 MB
  float* Y1 = (float*)d_ws;
  float* Y2 = Y1 + ysz;

  dim3 gg(2048 / 128, 512 / 128, 16);  // (16,4,16)
  // Y1 = adj @ X
  cheby_gemm512<<<gg, 256, 0, stream>>>(adj, x, nullptr, Y1, 0);
  // Y2 = L2 @ X = 2*adj@Y1 - X   (Chebyshev recursion on features; no L2 matrix)
  cheby_gemm512<<<gg, 256, 0, stream>>>(adj, Y1, x, Y2, 1);

  dim3 gm(32768 / 128, 16);  // (256,16)
  cheby_mix<<<gm, 256, 0, stream>>>(x, Y1, Y2, W, bias, out);
}